// CenterLossLayer_65438121721922
// MI455X (gfx1250) — compile-verified
//
#include <hip/hip_runtime.h>

#define NB 4096    // batch
#define NC 10000   // classes
#define ND 256     // feature dim
#define ALPHA 0.5f

typedef __attribute__((ext_vector_type(2))) float v2f;
typedef __attribute__((ext_vector_type(8))) float v8f;

// ---------------------------------------------------------------------------
// Kernel 1: recover label[b] = argmax(onehot[b,:]). One block per row.
// This is the roofline pass (160 MB of onehot). float4 loads -> b128 vmem,
// 4x fewer load instructions, pure bandwidth-bound streaming.
// ---------------------------------------------------------------------------
__global__ void labels_kernel(const float* __restrict__ onehot,
                              int* __restrict__ labels) {
  const int b = blockIdx.x;
  const float4* row4 = (const float4*)(onehot + (size_t)b * NC);
  for (int c4 = threadIdx.x; c4 < NC / 4; c4 += blockDim.x) {
    float4 v = row4[c4];
    if (v.x > 0.5f) labels[b] = 4 * c4 + 0;
    if (v.y > 0.5f) labels[b] = 4 * c4 + 1;
    if (v.z > 0.5f) labels[b] = 4 * c4 + 2;
    if (v.w > 0.5f) labels[b] = 4 * c4 + 3;
  }
}

// ---------------------------------------------------------------------------
// Kernel 2: per-class center update, fully deterministic (no float atomics).
// new_centers[c] = centers[c] - ALPHA*(n_c*centers[c] - sum feat)/(n_c+1).
// One block per class; each thread keeps its 16-label chunk in registers
// (4x int4), counts, gets a stable offset via an LDS scan, compacts member
// indices in sorted order, then sums feature rows in that fixed order.
// ---------------------------------------------------------------------------
__global__ void centers_kernel(const float* __restrict__ features,
                               const float* __restrict__ centers,
                               const int* __restrict__ labels,
                               float* __restrict__ new_centers) {
  const int c = blockIdx.x;
  const int t = threadIdx.x;                 // 0..255, also the feature dim d
  __shared__ int s_cnt[256];
  __shared__ int s_mem[NB];                  // worst case: all rows same class

  // load my 16 contiguous labels once (b = t*16 .. t*16+15) as 4x int4
  const int4* lab4 = (const int4*)labels;
  int lab[16];
#pragma unroll
  for (int q = 0; q < 4; ++q) {
    int4 v = lab4[t * 4 + q];
    lab[4 * q + 0] = v.x;
    lab[4 * q + 1] = v.y;
    lab[4 * q + 2] = v.z;
    lab[4 * q + 3] = v.w;
  }

  // count matches in my chunk
  int m = 0;
#pragma unroll
  for (int i = 0; i < 16; ++i) m += (lab[i] == c);
  s_cnt[t] = m;
  __syncthreads();

  // stable exclusive offset + total (256-iteration serial scan: trivial cost)
  int off = 0, total = 0;
  for (int u = 0; u < 256; ++u) {
    int cu = s_cnt[u];
    total += cu;
    if (u < t) off += cu;
  }

  // compact member indices in increasing-b order (deterministic)
#pragma unroll
  for (int i = 0; i < 16; ++i) {
    if (lab[i] == c) s_mem[off++] = t * 16 + i;
  }
  __syncthreads();

  // ordered sum over class members; thread t owns dimension t (coalesced rows)
  float cen = centers[(size_t)c * ND + t];
  float sum = 0.0f;
  for (int i = 0; i < total; ++i) {
    sum += features[(size_t)s_mem[i] * ND + t];
  }
  float n = (float)total;
  float newc = cen - ALPHA * (n * cen - sum) / (n + 1.0f);
  new_centers[(size_t)c * ND + t] = newc;
}

// ---------------------------------------------------------------------------
// Kernel 3: loss[b] = sum_d (features[b,d] - centers[label[b],d])^2 via
// V_WMMA_F32_16X16X4_F32 (exact f32 math). One wave per 16-row tile: stage
// diff in LDS (float4), accumulate G = diff * diff^T over K=256 in 64 WMMA
// steps, read the diagonal. Stride 260 floats per row: row bases are 16B
// aligned (float4 stores legal) and the ds_load_2addr_b64 fragment reads are
// fully bank-conflict-free (banks 4m+k vs 4m+k+2, disjoint parity classes).
// A-fragment register image == required B-fragment (A^T) for the symmetric
// product, so the same v2f feeds SRC0 and SRC1.
// ---------------------------------------------------------------------------
__global__ void loss_kernel(const float* __restrict__ features,
                            const float* __restrict__ centers,
                            const int* __restrict__ labels,
                            float* __restrict__ loss) {
  constexpr int STR = 260;
  __shared__ float s_diff[16 * STR];
  __shared__ float s_tile[16 * 17];

  const int lane = threadIdx.x;              // 0..31, one full wave (EXEC all 1s)
  const int r0 = blockIdx.x * 16;

  // stage diff tile: 16 rows x 256 cols, float4 granularity
  for (int r = 0; r < 16; ++r) {
    const int lab = labels[r0 + r];
    const float4* frow4 = (const float4*)(features + (size_t)(r0 + r) * ND);
    const float4* crow4 = (const float4*)(centers + (size_t)lab * ND);
    float4* drow4 = (float4*)(&s_diff[r * STR]);
#pragma unroll
    for (int q = 0; q < 2; ++q) {
      const int cc4 = lane + 32 * q;         // 0..63
      float4 f = frow4[cc4];
      float4 cr = crow4[cc4];
      float4 d;
      d.x = f.x - cr.x; d.y = f.y - cr.y; d.z = f.z - cr.z; d.w = f.w - cr.w;
      drow4[cc4] = d;
    }
  }
  __syncthreads();

  const int m   = lane & 15;                 // matrix row for this lane
  const int off = (lane >> 4) * 2;           // lanes 16-31 carry K+2, K+3
  v8f acc = {};
  for (int k = 0; k < ND; k += 4) {
    v2f a;
    a.x = s_diff[m * STR + k + off];
    a.y = s_diff[m * STR + k + off + 1];
    // G += A * A^T  (A and B fragments share the same register image)
    acc = __builtin_amdgcn_wmma_f32_16x16x4_f32(
        /*neg_a=*/false, a, /*neg_b=*/false, a,
        /*c_mod=*/(short)0, acc, /*reuse_a=*/false, /*reuse_b=*/false);
  }

  // spill C/D tile (M = vgpr [+8 for upper lanes], N = lane&15), read diagonal
#pragma unroll
  for (int v = 0; v < 8; ++v) {
    const int M = (lane < 16) ? v : (v + 8);
    s_tile[M * 17 + m] = acc[v];
  }
  __syncthreads();
  if (lane < 16) loss[r0 + lane] = s_tile[lane * 17 + lane];
}

// ---------------------------------------------------------------------------
extern "C" void kernel_launch(void* const* d_in, const int* in_sizes, int n_in,
                              void* d_out, int out_size, void* d_ws, size_t ws_size,
                              hipStream_t stream) {
  const float* features = (const float*)d_in[0];  // [4096, 256]
  const float* onehot   = (const float*)d_in[1];  // [4096, 10000]
  const float* centers  = (const float*)d_in[2];  // [10000, 256]

  float* out        = (float*)d_out;
  float* loss       = out;          // [4096, 1]
  float* newcenters = out + NB;     // [10000, 256]

  int* labels = (int*)d_ws;         // 16 KB scratch

  labels_kernel <<<NB, 256, 0, stream>>>(onehot, labels);
  centers_kernel<<<NC, 256, 0, stream>>>(features, centers, labels, newcenters);
  loss_kernel   <<<NB / 16, 32, 0, stream>>>(features, centers, labels, loss);
}